// Retriever_77919296684601
// MI455X (gfx1250) — compile-verified
//
#include <hip/hip_runtime.h>
#include <hip/hip_bf16.h>
#include <limits.h>

typedef __attribute__((ext_vector_type(2))) float v2f;
typedef __attribute__((ext_vector_type(8))) float v8f;

// ---------------------------------------------------------------------------
// K0: init segment bookkeeping
// ---------------------------------------------------------------------------
__global__ void k0_init(int* __restrict__ seg_start, int* __restrict__ seg_cnt, int M) {
    int m = blockIdx.x * blockDim.x + threadIdx.x;
    if (m < M) { seg_start[m] = INT_MAX; seg_cnt[m] = 0; }
}

// ---------------------------------------------------------------------------
// Kpre: row-dot precompute. rows [0,N): start@W0 ; [N,2N): prompt@W1 ;
// [2N, 2N+M): neighbor@W3.  One wave (32 lanes) per row, coalesced loads,
// shuffle-xor reduction.
// ---------------------------------------------------------------------------
__global__ void kpre_rowdots(const float* __restrict__ start_emb,
                             const float* __restrict__ prompt_emb,
                             const float* __restrict__ neighbor_emb,
                             const float* __restrict__ W,
                             float* __restrict__ sW0, float* __restrict__ pW1,
                             float* __restrict__ nW3,
                             int N, int M, int D) {
    const int warp = threadIdx.x >> 5;
    const int lane = threadIdx.x & 31;
    const int rowId = blockIdx.x * (blockDim.x >> 5) + warp;
    const int totalRows = 2 * N + M;
    if (rowId >= totalRows) return;

    const float* src; const float* w; float* dst; int outIdx;
    if (rowId < N)            { src = start_emb   + (size_t)rowId * D;       w = W;         dst = sW0; outIdx = rowId; }
    else if (rowId < 2 * N)   { src = prompt_emb  + (size_t)(rowId - N) * D; w = W + D;     dst = pW1; outIdx = rowId - N; }
    else                      { src = neighbor_emb+ (size_t)(rowId - 2*N)*D; w = W + 3 * D; dst = nW3; outIdx = rowId - 2 * N; }

    float acc = 0.f;
    for (int d = lane; d < D; d += 32) acc += src[d] * w[d];
#pragma unroll
    for (int off = 16; off > 0; off >>= 1) acc += __shfl_xor(acc, off, 32);
    if (lane == 0) dst[outIdx] = acc;
}

// ---------------------------------------------------------------------------
// K1: relation pass.  One wave handles a tile of 15 relations.
//   A (16x4 f32)  = rel rows  (relation_emb ⊙ count_table gather), row15 = dup
//   B (4x16 f32)  = cols 0..14 gathered start_emb rows, col 15 = W2
//   D = A*B + D accumulated over K=768 -> D[m][m]=logit_m, D[m][15]=w2_m
// VGPR layouts per CDNA5 ISA §7.12.2: A vgpr0={K0|K2}, vgpr1={K1|K3} per lane
// half; B/C/D rows striped across lanes.  Extraction through per-wave LDS.
// ---------------------------------------------------------------------------
__global__ void __launch_bounds__(256)
k1_relation_wmma(const float* __restrict__ relation_emb,
                 const float* __restrict__ count_table,
                 const float* __restrict__ start_emb,
                 const float* __restrict__ W,
                 const int* __restrict__ counts,
                 const int* __restrict__ rel_start_idx,
                 const int* __restrict__ neighbor_idx,
                 float2* __restrict__ lw,          // [R] (logit, w2)
                 int* __restrict__ seg_start,
                 int* __restrict__ seg_cnt,
                 int R, int D, int numTiles) {
    __shared__ float lds[8 * 256];

    const int warp = threadIdx.x >> 5;
    const int lane = threadIdx.x & 31;
    const int tile = blockIdx.x * 8 + warp;
    if (tile >= numTiles) return;

    const int rb  = tile * 15;
    const int row = lane & 15;        // A row / B col handled by this lane
    const int kh  = lane >> 4;        // 0 -> K{0,1}, 1 -> K{2,3}

    // A-side relation for this lane's row (row 15 duplicates relation rb)
    int rA = rb + ((row < 15) ? row : 0);
    if (rA >= R) rA = R - 1;
    const float* relP = relation_emb + (size_t)rA * D;
    const float* ctP  = count_table  + (size_t)counts[rA] * D;

    // B-side pointer for this lane's column
    const float* bP;
    if (row < 15) {
        int rC = rb + row; if (rC >= R) rC = R - 1;
        bP = start_emb + (size_t)rel_start_idx[rC] * D;
    } else {
        bP = W + 2 * D;               // W2 segment
    }

    v8f acc = {0.f, 0.f, 0.f, 0.f, 0.f, 0.f, 0.f, 0.f};
    const int kBase = kh * 2;
#pragma unroll 4
    for (int kk = 0; kk < D; kk += 4) {
        const int d = kk + kBase;     // even -> float2 (8B) aligned
        const float2 av = *reinterpret_cast<const float2*>(relP + d);
        const float2 cv = *reinterpret_cast<const float2*>(ctP + d);
        const float2 bv = *reinterpret_cast<const float2*>(bP + d);
        v2f a = { av.x * cv.x, av.y * cv.y };
        v2f b = { bv.x, bv.y };
        acc = __builtin_amdgcn_wmma_f32_16x16x4_f32(
            /*neg_a=*/false, a, /*neg_b=*/false, b,
            /*c_mod=*/(short)0, acc, /*reuse_a=*/false, /*reuse_b=*/false);
    }

    // Spill D (16x16) to LDS in row-major: vgpr v, lanes0-15 -> (M=v, N=lane),
    // lanes16-31 -> (M=v+8, N=lane-16)
    float* wl = &lds[warp * 256];
#pragma unroll
    for (int v = 0; v < 8; ++v) {
        const int rr = v + (kh << 3);
        wl[rr * 16 + row] = acc[v];
    }
    // Same-wave LDS RAW: DS pipe is in-order; compiler inserts s_wait_dscnt.
    if (kh == 0 && row < 15) {
        const int r = rb + row;
        if (r < R) {
            float2 o;
            o.x = wl[row * 16 + row];   // logit_r
            o.y = wl[row * 16 + 15];    // dot(rel_r, W2)
            lw[r] = o;
            const int nid = neighbor_idx[r];
            atomicMin(&seg_start[nid], r);
            atomicAdd(&seg_cnt[nid], 1);
        }
    }
}

// ---------------------------------------------------------------------------
// K2: per-segment stable softmax-weighted sum of w2 (neighbor_idx sorted ->
// segment = contiguous range [start, start+cnt)).  Avg ~10 rel/segment.
// ---------------------------------------------------------------------------
__global__ void k2_segment(const float2* __restrict__ lw,
                           const int* __restrict__ seg_start,
                           const int* __restrict__ seg_cnt,
                           float* __restrict__ relC, int M) {
    int m = blockIdx.x * blockDim.x + threadIdx.x;
    if (m >= M) return;
    const int cn = seg_cnt[m];
    float out = 0.f;
    if (cn > 0) {
        const int st = seg_start[m];
        float mx = -3.402823466e+38f;
        for (int i = 0; i < cn; ++i) mx = fmaxf(mx, lw[st + i].x);
        float den = 0.f, num = 0.f;
        for (int i = 0; i < cn; ++i) {
            const float2 v = lw[st + i];
            const float e = expf(v.x - mx);
            den += e;
            num += e * v.y;
        }
        out = num / den;
    }
    relC[m] = out;
}

// ---------------------------------------------------------------------------
// K3: final combine per neighbor
// ---------------------------------------------------------------------------
__global__ void k3_final(const float* __restrict__ sW0, const float* __restrict__ pW1,
                         const float* __restrict__ nW3, const float* __restrict__ relC,
                         const int* __restrict__ start_idx, const float* __restrict__ b,
                         float* __restrict__ out, int M) {
    int m = blockIdx.x * blockDim.x + threadIdx.x;
    if (m >= M) return;
    const int s = start_idx[m];
    out[m] = sW0[s] + pW1[s] + relC[m] + nW3[m] + b[0];
}

// ---------------------------------------------------------------------------
extern "C" void kernel_launch(void* const* d_in, const int* in_sizes, int n_in,
                              void* d_out, int out_size, void* d_ws, size_t ws_size,
                              hipStream_t stream) {
    const float* start_emb    = (const float*)d_in[0];
    const float* prompt_emb   = (const float*)d_in[1];
    const float* relation_emb = (const float*)d_in[2];
    const float* neighbor_emb = (const float*)d_in[3];
    const float* count_table  = (const float*)d_in[4];
    const float* W            = (const float*)d_in[5];
    const float* b            = (const float*)d_in[6];
    const int*   counts        = (const int*)d_in[7];
    const int*   rel_start_idx = (const int*)d_in[8];
    const int*   neighbor_idx  = (const int*)d_in[9];
    const int*   start_idx     = (const int*)d_in[10];

    const int D = in_sizes[5] / 4;       // W is [1, 4D]
    const int N = in_sizes[0] / D;
    const int R = in_sizes[7];
    const int M = in_sizes[10];

    // workspace carve-out (16B aligned)
    char* ws = (char*)d_ws;
    size_t off = 0;
    auto carve = [&](size_t bytes) { char* p = ws + off; off = (off + bytes + 15) & ~(size_t)15; return p; };
    float2* lw       = (float2*)carve((size_t)R * sizeof(float2));
    int*    segStart = (int*)   carve((size_t)M * sizeof(int));
    int*    segCnt   = (int*)   carve((size_t)M * sizeof(int));
    float*  relC     = (float*) carve((size_t)M * sizeof(float));
    float*  sW0      = (float*) carve((size_t)N * sizeof(float));
    float*  pW1      = (float*) carve((size_t)N * sizeof(float));
    float*  nW3      = (float*) carve((size_t)M * sizeof(float));
    (void)ws_size;

    const int numTiles = (R + 14) / 15;

    k0_init<<<(M + 255) / 256, 256, 0, stream>>>(segStart, segCnt, M);

    const int rows = 2 * N + M;                 // wave-per-row
    kpre_rowdots<<<(rows + 7) / 8, 256, 0, stream>>>(
        start_emb, prompt_emb, neighbor_emb, W, sW0, pW1, nW3, N, M, D);

    k1_relation_wmma<<<(numTiles + 7) / 8, 256, 0, stream>>>(
        relation_emb, count_table, start_emb, W,
        counts, rel_start_idx, neighbor_idx,
        lw, segStart, segCnt, R, D, numTiles);

    k2_segment<<<(M + 255) / 256, 256, 0, stream>>>(lw, segStart, segCnt, relC, M);

    k3_final<<<(M + 255) / 256, 256, 0, stream>>>(
        sW0, pW1, nW3, relC, start_idx, b, (float*)d_out, M);
}